// glocapBlockFuzzy_42631845380371
// MI455X (gfx1250) — compile-verified
//
#include <hip/hip_runtime.h>
#include <hip/hip_bf16.h>
#include <math.h>

// Problem constants (from reference)
#define NB   64            // batch n
#define CIN  32            // input capsule channels
#define DD   16            // pose dims per input channel (= P*P)
#define WH   64            // spatial positions w*h
#define NCLS 20            // classes
#define CAPS 2048          // CIN*WH capsules
#define PD   16            // output pose dims (i*4+d)
// V workspace: NB*NCLS*CAPS*PD fp32 = 167.8 MB -> fits entirely in MI455X 192MB L2.

typedef float v2f __attribute__((ext_vector_type(2)));
typedef float v8f __attribute__((ext_vector_type(8)));

// ---------------------------------------------------------------------------
// Kernel A: V[n,k,cap,dim] = sum_j lp[n,c,wh,i,j] * weight[k,c,j,d]
// One workgroup per (c, n); per (n,c) it is a 256x4 @ 4x80 fp32 matmul done
// with V_WMMA_F32_16X16X4_F32 (the native CDNA5 fp32 K=4 shape): 16 row
// strips x 5 col tiles = 80 WMMAs per WG, one WMMA per output tile (no K loop).
// ---------------------------------------------------------------------------
__global__ void __launch_bounds__(256)
compute_V_wmma(const float* __restrict__ l,
               const float* __restrict__ wgt,
               float* __restrict__ V) {
    const int c = blockIdx.x;     // 0..31
    const int n = blockIdx.y;     // 0..63
    __shared__ float Als[256 * 4];   // A: [row=wh*4+i][j]
    __shared__ float Bls[4 * 80];    // B: [j][col=k*4+d]
    const int tid = threadIdx.x;

    // Stage A from l[n][c][dd][wh] (coalesced global reads).
    // lp[n,c,wh,i,j] = l[n,c, i*4+j, wh]
    const float* lsrc = l + ((size_t)n * CIN + c) * (DD * WH);
    for (int t = tid; t < DD * WH; t += 256) {
        int dd = t >> 6, wh = t & 63;
        int i = dd >> 2, j = dd & 3;
        Als[(wh * 4 + i) * 4 + j] = lsrc[t];
    }
    // Stage B from weight[k][c][j][d]; flat = k*512 + c*16 + j*4 + d
    for (int t = tid; t < 4 * 80; t += 256) {
        int j = t / 80, col = t % 80;
        int k = col >> 2, d = col & 3;
        Bls[j * 80 + col] = wgt[k * 512 + c * 16 + j * 4 + d];
    }
    __syncthreads();

    const int wave = tid >> 5, lane = tid & 31;
    const int lh = lane & 15, hi = lane >> 4;
    const int jb = hi * 2;                       // K base: lanes<16 -> K0/K1, else K2/K3

    for (int s = wave; s < 16; s += 8) {         // 16 row strips of 16, 8 waves
        // A fragment (16x4 f32, ISA 7.12.2 layout)
        const int Ma = s * 16 + lh;
        v2f a;
        a.x = Als[Ma * 4 + jb];
        a.y = Als[Ma * 4 + jb + 1];
        #pragma unroll
        for (int t = 0; t < 5; ++t) {            // 5 col tiles of 16 (80 = 20k x 4d)
            const int N = t * 16 + lh;
            v2f b;
            b.x = Bls[jb * 80 + N];
            b.y = Bls[(jb + 1) * 80 + N];
            v8f acc = {};
            acc = __builtin_amdgcn_wmma_f32_16x16x4_f32(
                false, a, false, b, (short)0, acc, false, false);
            // Scatter D fragment: lane holds D[M=s*16+v+hi*8][N], v=0..7
            const int k = N >> 2, d = N & 3;
            #pragma unroll
            for (int v = 0; v < 8; ++v) {
                const int Mr = s * 16 + v + hi * 8;
                const int wh = Mr >> 2, i = Mr & 3;
                const size_t idx =
                    (((size_t)n * NCLS + k) * CAPS + (c * WH + wh)) * PD + i * 4 + d;
                V[idx] = acc[v];
            }
        }
    }
}

// ---------------------------------------------------------------------------
// Kernel B: fuzzy routing. One WG per n (1024 threads = 32 wave32s).
// lane = class k (lanes 20..31 masked); each wave owns 64 capsules.
// g_new[k] = (sum_cap r*V) / (sum_cap r)  -> coeff never materialized.
// V is L2-resident (168MB < 192MB), streamed 4x.
// ---------------------------------------------------------------------------
__device__ __forceinline__ float wave_sum32(float v) {
    #pragma unroll
    for (int m = 16; m >= 1; m >>= 1) v += __shfl_xor(v, m, 32);
    return v;
}

__global__ void __launch_bounds__(1024)
routing_kernel(const float* __restrict__ V,
               const float* __restrict__ g_in,
               const float* __restrict__ beta_a,
               float* __restrict__ out) {
    const int n    = blockIdx.x;
    const int tid  = threadIdx.x;
    const int lane = tid & 31;
    const int wave = tid >> 5;
    const int k    = lane;                       // class id for lanes < NCLS

    __shared__ float gA[NCLS * PD];              // current g
    __shared__ float gP[NCLS * PD];              // g used by last routing iter (g_2)
    __shared__ float G[NCLS * PD];               // unnormalized mean accumulator
    __shared__ float S[NCLS];                    // sum of r
    __shared__ float numS[NCLS], denS[NCLS];     // sigma accumulators

    if (tid < NCLS * PD) gA[tid] = g_in[n * NCLS * PD + tid];
    __syncthreads();

    const float* Vn = V + (size_t)n * NCLS * CAPS * PD;
    const int cap0 = wave * 64, cap1 = cap0 + 64;

    for (int it = 0; it < 3; ++it) {
        if (tid < NCLS * PD) G[tid] = 0.0f;
        if (tid < NCLS)      S[tid] = 0.0f;
        __syncthreads();

        float Gl[PD];
        #pragma unroll
        for (int d = 0; d < PD; ++d) Gl[d] = 0.0f;
        float Sl = 0.0f;

        for (int cap = cap0; cap < cap1; ++cap) {
            float vr[PD];
            float invd = 0.0f;
            if (k < NCLS) {
                const float4* p =
                    (const float4*)(Vn + ((size_t)k * CAPS + cap) * PD);
                if (cap + 1 < cap1)
                    __builtin_prefetch(p + 4, 0, 0);   // global_prefetch next row
                float4 q0 = p[0], q1 = p[1], q2 = p[2], q3 = p[3];
                vr[0]=q0.x; vr[1]=q0.y; vr[2]=q0.z; vr[3]=q0.w;
                vr[4]=q1.x; vr[5]=q1.y; vr[6]=q1.z; vr[7]=q1.w;
                vr[8]=q2.x; vr[9]=q2.y; vr[10]=q2.z; vr[11]=q2.w;
                vr[12]=q3.x; vr[13]=q3.y; vr[14]=q3.z; vr[15]=q3.w;
                float dist = 0.0f;
                #pragma unroll
                for (int d = 0; d < PD; ++d) {
                    float df = vr[d] - gA[k * PD + d];
                    dist = fmaf(df, df, dist);
                }
                invd = 1.0f / dist;              // r_n^-1 (expo == 2)
            }
            const float rd = wave_sum32(invd);   // sum over classes
            if (k < NCLS) {
                float r = invd / rd;             // (1/(r_n*r_d))
                r = r * r;                       // ^ M_FUZZ
                Sl += r;
                #pragma unroll
                for (int d = 0; d < PD; ++d) Gl[d] = fmaf(r, vr[d], Gl[d]);
            }
        }
        if (k < NCLS) {
            atomicAdd(&S[k], Sl);
            #pragma unroll
            for (int d = 0; d < PD; ++d) atomicAdd(&G[k * PD + d], Gl[d]);
        }
        __syncthreads();
        if (tid < NCLS * PD) {
            if (it == 2) gP[tid] = gA[tid];      // save g_2 for sigma pass
            gA[tid] = G[tid] / S[tid >> 4];      // g update
        }
        __syncthreads();
    }

    // Sigma pass: r from g_2 distances, spread measured against g_3.
    if (tid < NCLS) { numS[tid] = 0.0f; denS[tid] = 0.0f; }
    __syncthreads();
    float numl = 0.0f, denl = 0.0f;
    for (int cap = cap0; cap < cap1; ++cap) {
        float invd = 0.0f, d3 = 0.0f;
        if (k < NCLS) {
            const float4* p = (const float4*)(Vn + ((size_t)k * CAPS + cap) * PD);
            float4 q0 = p[0], q1 = p[1], q2 = p[2], q3 = p[3];
            float vr[PD];
            vr[0]=q0.x; vr[1]=q0.y; vr[2]=q0.z; vr[3]=q0.w;
            vr[4]=q1.x; vr[5]=q1.y; vr[6]=q1.z; vr[7]=q1.w;
            vr[8]=q2.x; vr[9]=q2.y; vr[10]=q2.z; vr[11]=q2.w;
            vr[12]=q3.x; vr[13]=q3.y; vr[14]=q3.z; vr[15]=q3.w;
            float d2 = 0.0f;
            #pragma unroll
            for (int d = 0; d < PD; ++d) {
                float a2 = vr[d] - gP[k * PD + d];
                float a3 = vr[d] - gA[k * PD + d];
                d2 = fmaf(a2, a2, d2);
                d3 = fmaf(a3, a3, d3);
            }
            invd = 1.0f / d2;
        }
        const float rd = wave_sum32(invd);
        if (k < NCLS) {
            float r = invd / rd;
            r = r * r;
            numl = fmaf(r, d3, numl);
            denl += r;
        }
    }
    if (k < NCLS) {
        atomicAdd(&numS[k], numl);
        atomicAdd(&denS[k], denl);
    }
    __syncthreads();

    if (tid < NCLS) {
        const float sigma_sq = numS[tid] / denS[tid];
        const float x = beta_a[tid] - 0.5f * logf(sigma_sq);   // LAMBDA = 1
        out[n * NCLS + tid] = 1.0f / (1.0f + __expf(-x));      // a -> out[0:1280)
    }
    if (tid < NCLS * PD)
        out[NB * NCLS + n * NCLS * PD + tid] = gA[tid];        // g -> out[1280:)
}

// ---------------------------------------------------------------------------
extern "C" void kernel_launch(void* const* d_in, const int* in_sizes, int n_in,
                              void* d_out, int out_size, void* d_ws, size_t ws_size,
                              hipStream_t stream) {
    const float* l      = (const float*)d_in[0];   // (64,32,16,8,8)
    const float* g      = (const float*)d_in[1];   // (64,20,16)
    const float* weight = (const float*)d_in[2];   // (20,32,4,4)
    const float* beta_a = (const float*)d_in[3];   // (20,1)
    float* out = (float*)d_out;                    // 1280 a + 20480 g
    float* V   = (float*)d_ws;                     // 167.8 MB, L2-resident

    (void)in_sizes; (void)n_in; (void)out_size; (void)ws_size;

    dim3 gridA(CIN, NB);
    compute_V_wmma<<<gridA, 256, 0, stream>>>(l, weight, V);
    routing_kernel<<<NB, 1024, 0, stream>>>(V, g, beta_a, out);
}